// FBI_RNN_74869869904096
// MI455X (gfx1250) — compile-verified
//
#include <hip/hip_runtime.h>
#include <hip/hip_bf16.h>

// ---------------------------------------------------------------------------
// Fused FBI-RNN forward for MI455X (gfx1250, wave32, WMMA).
//
//   ff  = x @ W_in^T                     (4096x1024x1024, fp32)
//   5x:  nsE = ff + aF @ W_fb^T ; nsF = aE @ W_ff^T   (TAU == 1)
//        aE  = softplus_b4(nsE) ; aF  = softplus_b4(nsF)
//   out = aE
//
// One workgroup (256 thr = 8 waves) owns a 16-row block and runs the entire
// recurrence in LDS; all GEMMs use V_WMMA_F32_16X16X4_F32 (full fp32
// precision — problem is memory-bound at 23.3 TB/s, ~30 GFLOP total, so
// lower precision buys nothing). Softplus uses the hardware transcendentals
// (v_exp_f32 / v_log_f32, TRANS32 — co-issues with VALU) in the numerically
// stable max(z,0) + log(1+exp(-|z|)) form.
// ---------------------------------------------------------------------------

typedef float v2f __attribute__((ext_vector_type(2)));
typedef float v8f __attribute__((ext_vector_type(8)));

#define B_ROWS   4096
#define IN_DIM   1024
#define OUT_DIM  1024
#define FBI_DIM  256
#define STEPS    5

#define ROWS     16          // rows per workgroup (one WMMA M-tile)
#define THREADS  256         // 8 waves of 32
#define WAVES    8
#define ETILES   8           // 64 output col-tiles / 8 waves
#define FTILES   2           // 16 FBI  col-tiles / 8 waves

// padded LDS strides (pad 4 floats -> bank = 4*m + k, conflict-free)
#define LDA      (OUT_DIM + 4)   // 1028, for ff and aE (also x staging)
#define LDF      (FBI_DIM + 4)   // 260,  for aF

#define SM_FF_OFF  0
#define SM_AE_OFF  (ROWS * LDA)
#define SM_AF_OFF  (2 * ROWS * LDA)
#define SM_TOTAL   (2 * ROWS * LDA + ROWS * LDF)   // 37056 floats = 144.8 KB

__device__ __forceinline__ float softplus_b4(float x) {
    // log(1 + exp(4x)) / 4, numerically stable; hardware exp/log (TRANS32)
    float z = 4.0f * x;
    float t = __expf(-fabsf(z));               // v_exp_f32 path
    return (fmaxf(z, 0.0f) + __logf(1.0f + t)) * 0.25f;  // v_log_f32 path
}

__device__ __forceinline__ v8f wmma_f32_k4(v2f a, v2f b, v8f c) {
    return __builtin_amdgcn_wmma_f32_16x16x4_f32(
        /*neg_a=*/false, a, /*neg_b=*/false, b,
        /*c_mod=*/(short)0, c, /*reuse_a=*/false, /*reuse_b=*/false);
}

__global__ __launch_bounds__(THREADS)
void fbi_rnn_fused(const float* __restrict__ x,
                   const float* __restrict__ W_in,
                   const float* __restrict__ W_fb,
                   const float* __restrict__ W_ff,
                   float* __restrict__ out) {
    __shared__ float smem[SM_TOTAL];
    float* sFF = smem + SM_FF_OFF;   // x tile during phase 1, then ff
    float* sAE = smem + SM_AE_OFF;
    float* sAF = smem + SM_AF_OFF;

    const int tid   = threadIdx.x;
    const int wave  = tid >> 5;
    const int lane  = tid & 31;
    const int mm    = lane & 15;          // A: row in tile / B,C: col in tile
    const int kb    = (lane >> 4) * 2;    // A/B: K sub-offset (0 or 2)
    const int chalf = (lane >> 4) * 8;    // C: row offset (0 or 8)
    const int row0  = blockIdx.x * ROWS;

    // ---------------- Phase 1: stage x tile into LDS ----------------
    for (int idx = tid; idx < ROWS * IN_DIM; idx += THREADS) {
        int m = idx >> 10, k = idx & (IN_DIM - 1);
        sFF[m * LDA + k] = x[(size_t)(row0 + m) * IN_DIM + k];
    }
    __syncthreads();

    // ff tile: each wave owns 8 consecutive 16-col tiles (128 cols)
    const int ecol0 = wave * (ETILES * 16);
    v8f cE[ETILES];
    #pragma unroll
    for (int t = 0; t < ETILES; ++t) cE[t] = (v8f){0,0,0,0,0,0,0,0};

    for (int k = 0; k < IN_DIM; k += 4) {
        v2f a = *(const v2f*)&sFF[mm * LDA + k + kb];
        #pragma unroll
        for (int t = 0; t < ETILES; ++t) {
            int n = ecol0 + t * 16 + mm;
            v2f b = *(const v2f*)&W_in[(size_t)n * IN_DIM + k + kb];
            cE[t] = wmma_f32_k4(a, b, cE[t]);
        }
    }
    __syncthreads();   // everyone done reading the x tile

    // scatter ff (C layout) into region 0, zero aE / aF
    #pragma unroll
    for (int t = 0; t < ETILES; ++t) {
        int n = ecol0 + t * 16 + mm;
        #pragma unroll
        for (int r = 0; r < 8; ++r)
            sFF[(chalf + r) * LDA + n] = cE[t][r];
    }
    for (int idx = tid; idx < ROWS * LDA; idx += THREADS) sAE[idx] = 0.0f;
    for (int idx = tid; idx < ROWS * LDF; idx += THREADS) sAF[idx] = 0.0f;
    __syncthreads();

    // ---------------- Phase 2: 5 synchronous settling steps ----------------
    const int fcol0 = wave * (FTILES * 16);
    for (int step = 0; step < STEPS; ++step) {
        // nsF = aE_prev @ W_ff^T    [16 x 256]
        v8f cF[FTILES];
        #pragma unroll
        for (int t = 0; t < FTILES; ++t) cF[t] = (v8f){0,0,0,0,0,0,0,0};
        for (int k = 0; k < OUT_DIM; k += 4) {
            v2f a = *(const v2f*)&sAE[mm * LDA + k + kb];
            #pragma unroll
            for (int t = 0; t < FTILES; ++t) {
                int n = fcol0 + t * 16 + mm;
                v2f b = *(const v2f*)&W_ff[(size_t)n * OUT_DIM + k + kb];
                cF[t] = wmma_f32_k4(a, b, cF[t]);
            }
        }

        // nsE = ff + aF_prev @ W_fb^T    [16 x 1024], C seeded from ff
        v8f cE2[ETILES];
        #pragma unroll
        for (int t = 0; t < ETILES; ++t) {
            int n = ecol0 + t * 16 + mm;
            #pragma unroll
            for (int r = 0; r < 8; ++r)
                cE2[t][r] = sFF[(chalf + r) * LDA + n];
        }
        for (int k = 0; k < FBI_DIM; k += 4) {
            v2f a = *(const v2f*)&sAF[mm * LDF + k + kb];
            #pragma unroll
            for (int t = 0; t < ETILES; ++t) {
                int n = ecol0 + t * 16 + mm;
                v2f b = *(const v2f*)&W_fb[(size_t)n * FBI_DIM + k + kb];
                cE2[t] = wmma_f32_k4(a, b, cE2[t]);
            }
        }
        __syncthreads();   // all waves done reading aE_prev / aF_prev

        // synchronous activity update
        #pragma unroll
        for (int t = 0; t < ETILES; ++t) {
            int n = ecol0 + t * 16 + mm;
            #pragma unroll
            for (int r = 0; r < 8; ++r)
                sAE[(chalf + r) * LDA + n] = softplus_b4(cE2[t][r]);
        }
        #pragma unroll
        for (int t = 0; t < FTILES; ++t) {
            int n = fcol0 + t * 16 + mm;
            #pragma unroll
            for (int r = 0; r < 8; ++r)
                sAF[(chalf + r) * LDF + n] = softplus_b4(cF[t][r]);
        }
        __syncthreads();
    }

    // ---------------- Output: aE after settling ----------------
    for (int idx = tid; idx < ROWS * OUT_DIM; idx += THREADS) {
        int m = idx >> 10, o = idx & (OUT_DIM - 1);
        out[(size_t)(row0 + m) * OUT_DIM + o] = sAE[m * LDA + o];
    }
}

extern "C" void kernel_launch(void* const* d_in, const int* in_sizes, int n_in,
                              void* d_out, int out_size, void* d_ws, size_t ws_size,
                              hipStream_t stream) {
    (void)in_sizes; (void)n_in; (void)d_ws; (void)ws_size; (void)out_size;
    const float* x    = (const float*)d_in[0];   // [4096, 1024]
    const float* W_in = (const float*)d_in[1];   // [1024, 1024]
    const float* W_fb = (const float*)d_in[2];   // [1024, 256]
    const float* W_ff = (const float*)d_in[3];   // [256, 1024]
    float* out        = (float*)d_out;           // [4096, 1024]

    dim3 grid(B_ROWS / ROWS);   // 256 workgroups
    dim3 block(THREADS);        // 8 waves (wave32)
    fbi_rnn_fused<<<grid, block, 0, stream>>>(x, W_in, W_fb, W_ff, out);
}